// QuantizedStatefulRecurrent_33698313404693
// MI455X (gfx1250) — compile-verified
//
#include <hip/hip_runtime.h>
#include <stdint.h>

// Problem constants (reference: B=8, T=4096, C=512).
#define B_ 8
#define T_ 4096
#define C_ 512
#define L_ 32              // timesteps per chunk
#define LOG2L_ 5           // log2(L_)
#define K_ (T_ / L_)       // 128 chunks
#define THREADS_ (C_ / 4)  // 128 threads/block, 4 channels (one float4) each
#define WS_NEEDED_ ((size_t)B_ * K_ * C_ * 2 * sizeof(float))  // 4 MB

// Native clang vector type (works with nontemporal builtins, unlike HIP float4).
typedef float v4f __attribute__((ext_vector_type(4)));

// ---------------------------------------------------------------------------
// CDNA5 async global->LDS staging (tracked by ASYNCcnt).
// Each lane stages 16B of the current input row into LDS; it consumes only its
// own bytes, so no workgroup barrier is required — ASYNCcnt ordering suffices.
// ---------------------------------------------------------------------------
__device__ __forceinline__ void async_ld_b128(unsigned lds_off, unsigned voff,
                                              uint64_t sbase) {
  asm volatile("global_load_async_to_lds_b128 %0, %1, %2 offset:0"
               :
               : "v"(lds_off), "v"(voff), "s"(sbase)
               : "memory");
}

// ---------------------------------------------------------------------------
// Kernel 1: per-chunk local scan with zero initial state; store chunk-final
// (real,imag) state to ws[b][k][c][2].
// ---------------------------------------------------------------------------
__global__ __launch_bounds__(THREADS_) void k_chunk_tails(
    const float* __restrict__ in, const float* __restrict__ Ar,
    const float* __restrict__ Ai, float* __restrict__ ws) {
  const int k = blockIdx.x;   // chunk index
  const int b = blockIdx.y;   // batch index
  const int c0 = threadIdx.x * 4;

  const v4f a4r = *(const v4f*)(Ar + c0);
  const v4f a4i = *(const v4f*)(Ai + c0);
  const float ar[4] = {a4r.x, a4r.y, a4r.z, a4r.w};
  const float ai[4] = {a4i.x, a4i.y, a4i.z, a4i.w};

  float sr[4] = {0.f, 0.f, 0.f, 0.f};
  float si[4] = {0.f, 0.f, 0.f, 0.f};

  const float* row = in + ((size_t)b * T_ + (size_t)k * L_) * C_ + c0;
#pragma unroll 4
  for (int t = 0; t < L_; ++t) {
    const v4f xv = *(const v4f*)row;
    row += C_;
    const float x[4] = {xv.x, xv.y, xv.z, xv.w};
#pragma unroll
    for (int j = 0; j < 4; ++j) {
      const float nr = fmaf(sr[j], ar[j], fmaf(-si[j], ai[j], x[j]));
      const float ni = fmaf(sr[j], ai[j], si[j] * ar[j]);
      sr[j] = nr;
      si[j] = ni;
    }
  }

  float* w = ws + (((size_t)b * K_ + k) * C_ + c0) * 2;
  *(v4f*)(w)     = (v4f){sr[0], si[0], sr[1], si[1]};
  *(v4f*)(w + 4) = (v4f){sr[2], si[2], sr[3], si[3]};
}

// ---------------------------------------------------------------------------
// Kernel 2: exclusive scan of chunk carries per (b,c):
//   carry_0 = 0 ; carry_j = A^L * carry_{j-1} + tail_{j-1}
// Overwrites ws in place with the carry (state entering each chunk).
// ---------------------------------------------------------------------------
__global__ __launch_bounds__(256) void k_carries(const float* __restrict__ Ar,
                                                 const float* __restrict__ Ai,
                                                 float* __restrict__ ws) {
  const int idx = blockIdx.x * 256 + threadIdx.x;  // 0 .. B_*C_-1
  const int b = idx / C_;
  const int c = idx % C_;

  // A^L via repeated squaring (L_ = 2^LOG2L_).
  float pr = Ar[c], pi = Ai[c];
#pragma unroll
  for (int s = 0; s < LOG2L_; ++s) {
    const float nr = pr * pr - pi * pi;
    const float ni = 2.0f * pr * pi;
    pr = nr;
    pi = ni;
  }

  float cr = 0.f, ci = 0.f;
  float* w = ws + ((size_t)b * K_ * C_ + c) * 2;
#pragma unroll 4
  for (int j = 0; j < K_; ++j) {
    const float er = w[0];
    const float ei = w[1];
    w[0] = cr;  // exclusive carry for chunk j
    w[1] = ci;
    const float nr = fmaf(pr, cr, fmaf(-pi, ci, er));
    const float ni = fmaf(pr, ci, fmaf(pi, cr, ei));
    cr = nr;
    ci = ni;
    w += (size_t)C_ * 2;
  }
}

// ---------------------------------------------------------------------------
// Kernel 3: seed each chunk with its carry and emit all states.
// Input rows are staged through LDS with a 4-deep async pipeline; outputs are
// interleaved (r,i) float4 non-temporal stores (never re-read -> keep L2 for
// the input stream).
// ---------------------------------------------------------------------------
__global__ __launch_bounds__(THREADS_) void k_final(
    const float* __restrict__ in, const float* __restrict__ Ar,
    const float* __restrict__ Ai, const float* __restrict__ ws,
    float* __restrict__ out) {
  __shared__ __align__(16) float smem[4][C_];

  const int k = blockIdx.x;
  const int b = blockIdx.y;
  const int c0 = threadIdx.x * 4;

  const uint64_t rowbase =
      (uint64_t)(uintptr_t)(in + ((size_t)b * T_ + (size_t)k * L_) * C_);
  const unsigned lane_goff = (unsigned)(c0 * 4);  // byte offset within a row
  unsigned lds_off[4];
#pragma unroll
  for (int p = 0; p < 4; ++p)
    lds_off[p] = (unsigned)(uintptr_t)(&smem[p][c0]);

  // Prologue: stage t = 0,1,2.
  async_ld_b128(lds_off[0], lane_goff + 0u * (C_ * 4u), rowbase);
  async_ld_b128(lds_off[1], lane_goff + 1u * (C_ * 4u), rowbase);
  async_ld_b128(lds_off[2], lane_goff + 2u * (C_ * 4u), rowbase);

  const v4f a4r = *(const v4f*)(Ar + c0);
  const v4f a4i = *(const v4f*)(Ai + c0);
  const float ar[4] = {a4r.x, a4r.y, a4r.z, a4r.w};
  const float ai[4] = {a4i.x, a4i.y, a4i.z, a4i.w};

  // Seed with the carry (state entering this chunk).
  const float* w = ws + (((size_t)b * K_ + k) * C_ + c0) * 2;
  const v4f w0 = *(const v4f*)(w);
  const v4f w1 = *(const v4f*)(w + 4);
  float sr[4] = {w0.x, w0.z, w1.x, w1.z};
  float si[4] = {w0.y, w0.w, w1.y, w1.w};

  float* orow = out + (((size_t)b * T_ + (size_t)k * L_) * C_ + c0) * 2;

  for (int t = 0; t < L_; ++t) {
    if (t + 3 < L_) {
      async_ld_b128(lds_off[(t + 3) & 3],
                    lane_goff + (unsigned)((t + 3) * (C_ * 4)), rowbase);
      // 4 outstanding after issue; wait until <=3 -> buffer t is complete.
      asm volatile("s_wait_asynccnt 0x3" ::: "memory");
    } else {
      // Tail: drain everything (slightly conservative, last 3 iters only).
      asm volatile("s_wait_asynccnt 0x0" ::: "memory");
    }

    const v4f xv = *(const v4f*)(&smem[t & 3][c0]);
    const float x[4] = {xv.x, xv.y, xv.z, xv.w};
#pragma unroll
    for (int j = 0; j < 4; ++j) {
      const float nr = fmaf(sr[j], ar[j], fmaf(-si[j], ai[j], x[j]));
      const float ni = fmaf(sr[j], ai[j], si[j] * ar[j]);
      sr[j] = nr;
      si[j] = ni;
    }

    __builtin_nontemporal_store((v4f){sr[0], si[0], sr[1], si[1]},
                                (v4f*)(orow));
    __builtin_nontemporal_store((v4f){sr[2], si[2], sr[3], si[3]},
                                (v4f*)(orow + 4));
    orow += 2 * C_;
  }
}

// ---------------------------------------------------------------------------
// Fallback: self-contained single-pass scan (used only if d_ws is too small).
// One thread per (b,c) chain; coalesced within a wave across c.
// ---------------------------------------------------------------------------
__global__ __launch_bounds__(256) void k_naive(const float* __restrict__ in,
                                               const float* __restrict__ Ar,
                                               const float* __restrict__ Ai,
                                               float* __restrict__ out) {
  const int idx = blockIdx.x * 256 + threadIdx.x;  // 0 .. B_*C_-1
  const int b = idx / C_;
  const int c = idx % C_;
  const float ar = Ar[c], ai = Ai[c];
  float sr = 0.f, si = 0.f;
  const float* p = in + (size_t)b * T_ * C_ + c;
  float* o = out + ((size_t)b * T_ * C_ + c) * 2;
#pragma unroll 4
  for (int t = 0; t < T_; ++t) {
    const float x = *p;
    p += C_;
    const float nr = fmaf(sr, ar, fmaf(-si, ai, x));
    const float ni = fmaf(sr, ai, si * ar);
    sr = nr;
    si = ni;
    o[0] = sr;
    o[1] = si;
    o += 2 * C_;
  }
}

// ---------------------------------------------------------------------------
// Host launcher. Inputs (setup_inputs order): inputs[B,T,C] f32,
// A_real[C] f32, A_imag[C] f32. Output: [B,T,C,2] f32.
// ---------------------------------------------------------------------------
extern "C" void kernel_launch(void* const* d_in, const int* in_sizes, int n_in,
                              void* d_out, int out_size, void* d_ws,
                              size_t ws_size, hipStream_t stream) {
  (void)in_sizes;
  (void)n_in;
  (void)out_size;
  const float* in = (const float*)d_in[0];
  const float* Ar = (const float*)d_in[1];
  const float* Ai = (const float*)d_in[2];
  float* out = (float*)d_out;
  float* ws = (float*)d_ws;

  if (ws_size >= WS_NEEDED_) {
    dim3 grid(K_, B_);
    k_chunk_tails<<<grid, THREADS_, 0, stream>>>(in, Ar, Ai, ws);
    k_carries<<<(B_ * C_) / 256, 256, 0, stream>>>(Ar, Ai, ws);
    k_final<<<grid, THREADS_, 0, stream>>>(in, Ar, Ai, ws, out);
  } else {
    // Deterministic fallback (no scratch needed).
    k_naive<<<(B_ * C_) / 256, 256, 0, stream>>>(in, Ar, Ai, out);
  }
}